// LGA3D3_15796889715115
// MI455X (gfx1250) — compile-verified
//
#include <hip/hip_runtime.h>

constexpr int C = 16, D = 48, H = 128, W = 256;
constexpr int DT = 24;            // depth outputs per workgroup (D/2)
constexpr int TD = DT + 2;        // tile depth rows including +-1 halo
constexpr int RS = 264;           // LDS row stride in floats; 264*4 = 66*16B (keeps rows 16B aligned)
                                  // col = w' + 4, w' in [-2,257] -> cols 2..261; interior w'=0 at col 4 (16B aligned)
constexpr int TILE_FLOATS = 5 * TD * RS;   // 34320 floats = 137,280 bytes of LDS

// 16-byte vector type matching the builtin's expected parameter:
//   'int __attribute__((__vector_size__(4 * sizeof(int)))) *'
typedef int v4i __attribute__((vector_size(4 * sizeof(int))));

__device__ __forceinline__ void async_copy16(const float* g, float* l) {
#if defined(__AMDGCN__) && __has_builtin(__builtin_amdgcn_global_load_async_to_lds_b128)
  // async DMA: 16 bytes per lane, tracked by ASYNCcnt (no VGPR staging)
  __builtin_amdgcn_global_load_async_to_lds_b128(
      (v4i*)g, (v4i*)l, /*imm offset*/0, /*cpol*/0);
#else
  float4 v = *(const float4*)g;   // both sides are 16B aligned by construction
  *(float4*)l = v;
#endif
}

__device__ __forceinline__ void wait_async() {
#if defined(__AMDGCN__) && __has_builtin(__builtin_amdgcn_s_wait_asynccnt)
  __builtin_amdgcn_s_wait_asynccnt(0);
#elif defined(__AMDGCN__)
  asm volatile("s_wait_asynccnt 0" ::: "memory");
#endif
}

__global__ __launch_bounds__(256) void lga3d_pass(const float* __restrict__ in,
                                                  const float* __restrict__ filt,
                                                  float* __restrict__ out) {
  __shared__ float tile[TILE_FLOATS];

  const int tid = threadIdx.x;        // 0..255  <-> output w
  const int h   = blockIdx.x;         // 0..127
  const int c   = blockIdx.y;         // 0..15
  const int d0  = blockIdx.z * DT;    // 0 or 24

  // 1) zero entire tile: makes depth halo, spatial halo, and OOB rows all
  //    contribute exactly zero -> branch-free compute below.
  for (int idx = tid; idx < TILE_FLOATS; idx += 256) tile[idx] = 0.0f;
  __syncthreads();   // zero-stores committed to LDS before async DMA writes

  // 2) async-stage the valid interior: rows h'=h+i-2 in range, depths d'=d0-1+t in [0,D).
  const int tlo = (d0 == 0) ? 1 : 0;            // half0: t=1..25, half1: t=0..24
  const int thi = (d0 + DT >= D) ? (TD - 2) : (TD - 1);
  const int nch = (thi - tlo + 1) * 64;         // 64 x 16B chunks per 256-float row
  for (int i = 0; i < 5; ++i) {
    const int hh = h + i - 2;
    if (hh < 0 || hh >= H) continue;            // uniform branch per workgroup
    for (int cidx = tid; cidx < nch; cidx += 256) {
      const int tr = tlo + (cidx >> 6);
      const int ck = cidx & 63;
      const int dd = d0 - 1 + tr;
      const float* g = in + ((c * D + dd) * H + hh) * W + (ck << 2);
      float* l = &tile[(i * TD + tr) * RS + 4 + (ck << 2)];
      async_copy16(g, l);
    }
  }
  wait_async();
  __syncthreads();

  // 3) branch-free FMA sweep: 75 taps x 24 depths per thread = 1800 FMAs,
  //    sharing each LDS value across the 3 depth-offset weights.
  float acc[DT];
#pragma unroll
  for (int d = 0; d < DT; ++d) acc[d] = 0.0f;

  const int hw = h * W + tid;
  for (int i = 0; i < 5; ++i) {
    const float* trow = &tile[i * TD * RS];
#pragma unroll
    for (int j = 0; j < 5; ++j) {
      const int f = i * 5 + j;
      const float w0 = filt[(f     ) * (H * W) + hw];   // depth offset -1 group (g=0)
      const float w1 = filt[(f + 25) * (H * W) + hw];   // g=1
      const float w2 = filt[(f + 50) * (H * W) + hw];   // g=2
      const int col = tid + j + 2;                       // w' = w+j-2 -> col w'+4
#pragma unroll
      for (int t = 0; t < TD; ++t) {
        const float v = trow[t * RS + col];              // stride-1 across lanes: no bank conflicts
        if (t < DT)                acc[t]     = fmaf(w0, v, acc[t]);
        if (t >= 1 && t - 1 < DT)  acc[t - 1] = fmaf(w1, v, acc[t - 1]);
        if (t >= 2)                acc[t - 2] = fmaf(w2, v, acc[t - 2]);
      }
    }
  }

  // 4) store: coalesced across w for each depth
  float* ob = out + ((c * D + d0) * H + h) * W + tid;
#pragma unroll
  for (int d = 0; d < DT; ++d) ob[d * (H * W)] = acc[d];
}

extern "C" void kernel_launch(void* const* d_in, const int* in_sizes, int n_in,
                              void* d_out, int out_size, void* d_ws, size_t ws_size,
                              hipStream_t stream) {
  const float* cost = (const float*)d_in[0];
  const float* filt = (const float*)d_in[1];
  float* out = (float*)d_out;
  float* ws0 = (float*)d_ws;
  const size_t N = (size_t)C * D * H * W;

  dim3 grid(H, C, D / DT);   // 128 x 16 x 2 = 4096 workgroups
  dim3 block(256);           // 8 wave32 waves

  if (ws_size >= 2 * N * sizeof(float)) {
    float* ws1 = ws0 + N;
    lga3d_pass<<<grid, block, 0, stream>>>(cost, filt, ws0);
    lga3d_pass<<<grid, block, 0, stream>>>(ws0,  filt, ws1);
    lga3d_pass<<<grid, block, 0, stream>>>(ws1,  filt, out);
  } else {
    // ping-pong through d_out when scratch only holds one tensor
    lga3d_pass<<<grid, block, 0, stream>>>(cost, filt, out);
    lga3d_pass<<<grid, block, 0, stream>>>(out,  filt, ws0);
    lga3d_pass<<<grid, block, 0, stream>>>(ws0,  filt, out);
  }
}